// RelationalDynamicAggregation_37263136260146
// MI455X (gfx1250) — compile-verified
//
#include <hip/hip_runtime.h>

typedef float v2f __attribute__((ext_vector_type(2)));
typedef float v4f __attribute__((ext_vector_type(4)));
typedef float v8f __attribute__((ext_vector_type(8)));

#define B_  512
#define T_  16
#define H_  200
#define HP  208      // H padded to multiple of 16
#define R_  16
#define V_  64

// LDS layout (float offsets)
#define SEQ_OFF   0
#define SEQT_OFF  (HP * V_)                 // 13312
#define RI_OFF    (SEQT_OFF + V_ * HP)      // 26624
#define MV_OFF    (RI_OFF + T_ * R_ * V_)   // 43008
#define SMEM_FLOATS (MV_OFF + 256)          // 43264 floats = 173056 B

__global__ __launch_bounds__(512, 1)
void rda_kernel(const float* __restrict__ seq,
                const float* __restrict__ target,
                const float* __restrict__ target_value,
                const float* __restrict__ r_emb,
                const unsigned char* __restrict__ valid_mask,
                float* __restrict__ context_out,
                float* __restrict__ ta_out)
{
    extern __shared__ float smem[];
    float* seq_s  = smem + SEQ_OFF;    // [HP][V] row-major
    float* seqT_s = smem + SEQT_OFF;   // [V][HP] (transposed copy for GEMM2 B)
    float* ri_s   = smem + RI_OFF;     // [T*R][V]
    float* mv_s   = smem + MV_OFF;     // maskval[h]: 0.0 valid, -inf invalid/pad

    const int b    = blockIdx.x;
    const int tid  = threadIdx.x;
    const int lane = tid & 31;
    const int wave = tid >> 5;    // wave w handles t = w
    const int half = lane >> 4;   // 0: lanes 0-15, 1: lanes 16-31
    const int ln   = lane & 15;

    // ---------------- Phase 0: stage LDS ----------------
    const float* seq_g = seq + (size_t)b * H_ * V_;
    for (int i4 = tid; i4 < HP * (V_ / 4); i4 += 512) {
        int h = i4 >> 4;
        int v = (i4 & 15) << 2;
        v4f val = {0.f, 0.f, 0.f, 0.f};
        if (h < H_) val = *(const v4f*)(seq_g + h * V_ + v);
        *(v4f*)(seq_s + h * V_ + v) = val;
        seqT_s[(v + 0) * HP + h] = val[0];
        seqT_s[(v + 1) * HP + h] = val[1];
        seqT_s[(v + 2) * HP + h] = val[2];
        seqT_s[(v + 3) * HP + h] = val[3];
    }
    const float* tv_g = target_value + (size_t)b * T_ * R_ * V_;
    const float* tg_g = target + (size_t)b * T_ * V_;
    for (int i4 = tid; i4 < T_ * R_ * (V_ / 4); i4 += 512) {
        int t = i4 >> 8;
        int r = (i4 >> 4) & 15;
        int v = (i4 & 15) << 2;
        v4f re = *(const v4f*)(r_emb + r * V_ + v);
        v4f tv = *(const v4f*)(tv_g + (t * R_ + r) * V_ + v);
        v4f tg = *(const v4f*)(tg_g + t * V_ + v);
        *(v4f*)(ri_s + (t * R_ + r) * V_ + v) = (re + tv) * tg;
    }
    for (int h = tid; h < 256; h += 512) {
        float m = -__builtin_inff();
        if (h < H_ && valid_mask[b * H_ + h]) m = 0.0f;
        mv_s[h] = m;
    }
    __syncthreads();

    // ---------------- Phase 1: GEMM1  attn[h, n] = seq . ri^T ----------------
    // B operand (ri columns for this wave) is invariant over m-tiles: hoist.
    v2f bop[16];
    {
        const float* rrow = ri_s + (wave * R_ + ln) * V_ + 2 * half;
        #pragma unroll
        for (int k = 0; k < 16; ++k)
            bop[k] = *(const v2f*)(rrow + 4 * k);
    }
    v8f acc1[13];
    #pragma unroll
    for (int mt = 0; mt < 13; ++mt) {
        const float* srow = seq_s + (mt * 16 + ln) * V_ + 2 * half;
        v8f c = {0.f, 0.f, 0.f, 0.f, 0.f, 0.f, 0.f, 0.f};
        #pragma unroll
        for (int k = 0; k < 16; ++k) {
            v2f a = *(const v2f*)(srow + 4 * k);
            c = __builtin_amdgcn_wmma_f32_16x16x4_f32(false, a, false, bop[k],
                                                      (short)0, c, false, false);
        }
        acc1[mt] = c;
    }

    // ---------------- Phase 2: ta (wave 0 == t 0): softmax over r per h ----
    if (wave == 0) {
        #pragma unroll
        for (int mt = 0; mt < 13; ++mt) {
            #pragma unroll
            for (int i = 0; i < 8; ++i) {
                float x = acc1[mt][i];
                float m = x;
                m = fmaxf(m, __shfl_xor(m, 1, 32));
                m = fmaxf(m, __shfl_xor(m, 2, 32));
                m = fmaxf(m, __shfl_xor(m, 4, 32));
                m = fmaxf(m, __shfl_xor(m, 8, 32));
                float e = __expf(x - m);
                float s = e;
                s += __shfl_xor(s, 1, 32);
                s += __shfl_xor(s, 2, 32);
                s += __shfl_xor(s, 4, 32);
                s += __shfl_xor(s, 8, 32);
                float taval = e / s;
                int h = mt * 16 + i + 8 * half;
                if (h < H_) {
                    if (mv_s[h] != 0.0f) taval = 0.0f;   // invalid h -> 0
                    ta_out[(size_t)b * H_ * R_ + h * R_ + ln] = taval;
                }
            }
        }
    }

    // ---------------- Phase 3: mask + softmax over h (per column n) --------
    float pmax = -__builtin_inff();
    #pragma unroll
    for (int mt = 0; mt < 13; ++mt) {
        const float* mp = mv_s + mt * 16 + 8 * half;
        v4f m0 = *(const v4f*)(mp);
        v4f m1 = *(const v4f*)(mp + 4);
        acc1[mt][0] += m0[0]; acc1[mt][1] += m0[1];
        acc1[mt][2] += m0[2]; acc1[mt][3] += m0[3];
        acc1[mt][4] += m1[0]; acc1[mt][5] += m1[1];
        acc1[mt][6] += m1[2]; acc1[mt][7] += m1[3];
        #pragma unroll
        for (int i = 0; i < 8; ++i) pmax = fmaxf(pmax, acc1[mt][i]);
    }
    float cmax = fmaxf(pmax, __shfl_xor(pmax, 16, 32));
    float psum = 0.f;
    #pragma unroll
    for (int mt = 0; mt < 13; ++mt) {
        #pragma unroll
        for (int i = 0; i < 8; ++i) {
            float e = __expf(acc1[mt][i] - cmax);  // -inf -> 0
            acc1[mt][i] = e;
            psum += e;
        }
    }
    float csum = psum + __shfl_xor(psum, 16, 32);
    float rinv = 1.0f / csum;                      // h=0 always valid -> csum > 0
    #pragma unroll
    for (int mt = 0; mt < 13; ++mt) {
        #pragma unroll
        for (int i = 0; i < 8; ++i) acc1[mt][i] *= rinv;
    }

    // ---------------- Phase 4: GEMM2  context[n, v] = w^T . seq ------------
    v8f acc2[4];
    #pragma unroll
    for (int vt = 0; vt < 4; ++vt)
        acc2[vt] = (v8f){0.f, 0.f, 0.f, 0.f, 0.f, 0.f, 0.f, 0.f};

    #pragma unroll
    for (int tile = 0; tile < 13; ++tile) {
        #pragma unroll
        for (int js = 0; js < 4; ++js) {
            const int j    = js * 4;       // h0 % 16
            const int base = j & 7;
            // Rebuild GEMM2 A operand (w[h0..h0+3, n]) from accumulator layout.
            float t0 = acc1[tile][base + 0];
            float t1 = acc1[tile][base + 1];
            float t2 = acc1[tile][base + 2];
            float t3 = acc1[tile][base + 3];
            float s0 = __shfl_xor(t0, 16, 32);
            float s1 = __shfl_xor(t1, 16, 32);
            float s2 = __shfl_xor(t2, 16, 32);
            float s3 = __shfl_xor(t3, 16, 32);
            v2f A;
            if (j < 8) {
                A[0] = (half == 0) ? t0 : s2;
                A[1] = (half == 0) ? t1 : s3;
            } else {
                A[0] = (half == 0) ? s0 : t2;
                A[1] = (half == 0) ? s1 : t3;
            }
            const int h0 = tile * 16 + j;
            #pragma unroll
            for (int vt = 0; vt < 4; ++vt) {
                v2f Bv = *(const v2f*)(seqT_s + (vt * 16 + ln) * HP + h0 + 2 * half);
                acc2[vt] = __builtin_amdgcn_wmma_f32_16x16x4_f32(false, A, false, Bv,
                                                                 (short)0, acc2[vt],
                                                                 false, false);
            }
        }
    }

    // ---------------- Phase 5: store context -------------------------------
    float* ctx = context_out + (size_t)(b * T_ + wave) * R_ * V_;
    #pragma unroll
    for (int vt = 0; vt < 4; ++vt) {
        #pragma unroll
        for (int i = 0; i < 8; ++i) {
            int r = i + 8 * half;
            ctx[r * V_ + vt * 16 + ln] = acc2[vt][i];
        }
    }
}

extern "C" void kernel_launch(void* const* d_in, const int* in_sizes, int n_in,
                              void* d_out, int out_size, void* d_ws, size_t ws_size,
                              hipStream_t stream) {
    const float*         seq          = (const float*)d_in[0];
    const float*         target       = (const float*)d_in[1];
    const float*         target_value = (const float*)d_in[2];
    const float*         r_emb        = (const float*)d_in[3];
    const unsigned char* valid_mask   = (const unsigned char*)d_in[4];

    float* ctx = (float*)d_out;                         // (B,T,R,V) first
    float* ta  = ctx + (size_t)B_ * T_ * R_ * V_;       // then (B,H,R)

    const size_t smem = (size_t)SMEM_FLOATS * sizeof(float);
    hipFuncSetAttribute(reinterpret_cast<const void*>(rda_kernel),
                        hipFuncAttributeMaxDynamicSharedMemorySize, (int)smem);
    rda_kernel<<<B_, 512, smem, stream>>>(seq, target, target_value, r_emb,
                                          valid_mask, ctx, ta);
}